// TopKMaxPooling_1580547968391
// MI455X (gfx1250) — compile-verified
//
#include <hip/hip_runtime.h>
#include <hip/hip_bf16.h>

// Problem constants (from reference): x is (64, 256, 64, 64) fp32.
#define HW      4096                  // h*w per (b,c) row
#define K_SEL   410                   // round(0.1 * 4096)
#define THREADS 256                   // 8 wave32s per block
#define VEC     4                     // floats per b128 transfer
#define CHUNK   (THREADS * VEC)       // 1024 floats per copy iteration
#define ITERS   (HW / CHUNK)          // 4

// Order-preserving float-bits -> uint key (larger float <=> larger uint key).
__device__ __forceinline__ unsigned f2k(unsigned u) {
    return u ^ (unsigned)(((int)u >> 31) | 0x80000000);
}
// Inverse transform: key -> float bits.
__device__ __forceinline__ float k2f(unsigned k) {
    unsigned bits = (k & 0x80000000u) ? (k ^ 0x80000000u) : ~k;
    return __uint_as_float(bits);
}

__global__ __launch_bounds__(THREADS)
void topk_sum_radix_kernel(const float* __restrict__ x, float* __restrict__ out) {
    __shared__ __align__(16) unsigned keys[HW];   // 16 KB of sortable keys
    __shared__ unsigned hist[THREADS];            // 256-bin digit histogram
    __shared__ unsigned bcast[2];                 // [0]=cut digit, [1]=new remaining
    __shared__ float    wsum[THREADS / 32];       // per-wave partial sums

    const int      tid  = threadIdx.x;
    const int      row  = blockIdx.x;             // 0 .. b*c-1
    const float*   rowp = x + (size_t)row * HW;

    // ---- Stage row into LDS via gfx1250 async tensor path (ASYNCcnt) ----
    // Each lane issues one 16-byte GLOBAL_LOAD_ASYNC_TO_LDS_B128 per iter.
    {
        unsigned lds_base = (unsigned)(size_t)(&keys[0]);   // low 32 bits = LDS offset
        #pragma unroll
        for (int i = 0; i < ITERS; ++i) {
            unsigned     elem = (unsigned)(i * CHUNK + tid * VEC);
            unsigned     loff = lds_base + elem * 4u;
            const float* gp   = rowp + elem;
            asm volatile("global_load_async_to_lds_b128 %0, %1, off"
                         :: "v"(loff), "v"(gp)
                         : "memory");
        }
    }
#if __has_builtin(__builtin_amdgcn_s_wait_asynccnt)
    __builtin_amdgcn_s_wait_asynccnt(0);
#else
    asm volatile("s_wait_asynccnt 0" ::: "memory");
#endif
    __syncthreads();

    // ---- In-place transform to monotone uint keys (b128 LDS traffic) ----
    {
        uint4* kv = (uint4*)keys;
        #pragma unroll
        for (int i = tid; i < HW / 4; i += THREADS) {
            uint4 v = kv[i];
            v.x = f2k(v.x); v.y = f2k(v.y); v.z = f2k(v.z); v.w = f2k(v.w);
            kv[i] = v;
        }
    }
    __syncthreads();

    // ---- MSB-first 8-bit radix select of the K_SEL-th largest key ----
    unsigned prefix    = 0;       // matched high bits so far
    unsigned remaining = K_SEL;   // how many still needed among candidates

    #pragma unroll
    for (int pass = 3; pass >= 0; --pass) {
        hist[tid] = 0;
        __syncthreads();

        const int      shift = pass * 8;
        const unsigned pmask = (pass == 3) ? 0u : (0xFFFFFFFFu << (shift + 8));

        // Histogram current digit among candidates (prefix match).
        {
            const uint4* kv = (const uint4*)keys;
            #pragma unroll
            for (int i = tid; i < HW / 4; i += THREADS) {
                uint4 v = kv[i];
                if ((v.x & pmask) == prefix) atomicAdd(&hist[(v.x >> shift) & 0xFF], 1u);
                if ((v.y & pmask) == prefix) atomicAdd(&hist[(v.y >> shift) & 0xFF], 1u);
                if ((v.z & pmask) == prefix) atomicAdd(&hist[(v.z >> shift) & 0xFF], 1u);
                if ((v.w & pmask) == prefix) atomicAdd(&hist[(v.w >> shift) & 0xFF], 1u);
            }
        }
        __syncthreads();

        // Inclusive suffix-sum over the 256 bins (Hillis-Steele, 8 steps).
        unsigned v = hist[tid];
        #pragma unroll
        for (int d = 1; d < THREADS; d <<= 1) {
            unsigned other = (tid + d < THREADS) ? hist[tid + d] : 0u;
            __syncthreads();
            v += other;
            hist[tid] = v;
            __syncthreads();
        }

        // Unique cut bin b: suffix[b] >= remaining > suffix[b+1].
        {
            unsigned suff     = v;
            unsigned suffNext = (tid < THREADS - 1) ? hist[tid + 1] : 0u;
            if (suff >= remaining && suffNext < remaining) {
                bcast[0] = (unsigned)tid;
                bcast[1] = remaining - suffNext;  // still needed inside this bin
            }
        }
        __syncthreads();
        prefix   |= bcast[0] << shift;
        remaining = bcast[1];
        __syncthreads();
    }

    // prefix == exact key of the K-th largest element; `remaining` = tie copies.
    const unsigned t = prefix;

    // ---- Sum pass: everything strictly above t, plus tie contribution ----
    float partial = 0.0f;
    {
        const uint4* kv = (const uint4*)keys;
        #pragma unroll
        for (int i = tid; i < HW / 4; i += THREADS) {
            uint4 v = kv[i];
            if (v.x > t) partial += k2f(v.x);
            if (v.y > t) partial += k2f(v.y);
            if (v.z > t) partial += k2f(v.z);
            if (v.w > t) partial += k2f(v.w);
        }
    }
    if (tid == 0) partial += (float)remaining * k2f(t);

    // ---- wave32 reduction then cross-wave reduction ----
    #pragma unroll
    for (int off = 16; off > 0; off >>= 1)
        partial += __shfl_down(partial, off, 32);
    if ((tid & 31) == 0) wsum[tid >> 5] = partial;
    __syncthreads();
    if (tid == 0) {
        float s = 0.0f;
        #pragma unroll
        for (int w = 0; w < THREADS / 32; ++w) s += wsum[w];
        out[row] = s * (1.0f / (float)K_SEL);
    }
}

extern "C" void kernel_launch(void* const* d_in, const int* in_sizes, int n_in,
                              void* d_out, int out_size, void* d_ws, size_t ws_size,
                              hipStream_t stream) {
    (void)n_in; (void)d_ws; (void)ws_size; (void)out_size;
    const float* x   = (const float*)d_in[0];
    float*       out = (float*)d_out;
    const int rows = in_sizes[0] / HW;   // 64 * 256 = 16384
    topk_sum_radix_kernel<<<rows, THREADS, 0, stream>>>(x, out);
}